// SparseMapMasking_29996051595886
// MI455X (gfx1250) — compile-verified
//
#include <hip/hip_runtime.h>

#define D 2048
#define K 8
#define NTHREADS 1024
#define NWAVES (NTHREADS / 32)

typedef __attribute__((ext_vector_type(2))) float v2f;
typedef __attribute__((ext_vector_type(4))) float v4f;
typedef __attribute__((ext_vector_type(4))) int   v4i;
typedef __attribute__((ext_vector_type(8))) float v8f;

// ---------------------------------------------------------------------------
// Kernel 1: cooperative single-workgroup active-set solver.
//   - th, g, ranks live in LDS (24KB of the 320KB WGP pool)
//   - V (8x2048 f32) and R (8x2048 i32) live in d_ws (128KB, L2-resident)
//   - Gram+b computed with V_WMMA_F32_16X16X4_F32 over a fused 16-row operand
// ---------------------------------------------------------------------------
__global__ __launch_bounds__(NTHREADS)
void sparse_rank_solver(const float* __restrict__ theta,
                        float* __restrict__ out,   // alphas at out[0..7]
                        float* __restrict__ Vg,    // K*D floats
                        int*   __restrict__ Rg)    // K*D ints
{
    __shared__ float th_s[D];
    __shared__ float g_s[D];
    __shared__ int   r_s[D];
    __shared__ float G_s[256];      // 16x16 Gram tile (row-major)
    __shared__ float q_s[K];
    __shared__ float a_s[K];
    __shared__ int   act_s[K];
    __shared__ int   flag_s;        // per-k "matches existing row" bits
    __shared__ int   add_s;

    const int tid = threadIdx.x;

    // ---- init ----
    for (int i = tid; i < D; i += NTHREADS) {
        th_s[i] = theta[i] / 1e-5f;           // theta[:,0] / TMP
        #pragma unroll
        for (int k = 0; k < K; ++k) { Vg[k * D + i] = 0.f; Rg[k * D + i] = -1; }
    }
    if (tid < K) { q_s[tid] = 0.f; a_s[tid] = 0.f; act_s[tid] = 0; }
    __threadfence();
    __syncthreads();

    for (int t = 0; t < K; ++t) {
        // ---- Phase A: g = th - q @ V ----
        for (int i = tid; i < D; i += NTHREADS) {
            float mu = 0.f;
            #pragma unroll
            for (int k = 0; k < K; ++k)
                if (act_s[k]) mu += q_s[k] * Vg[k * D + i];
            g_s[i] = th_s[i] - mu;
        }
        __syncthreads();

        // ---- Phase B: stable descending-argsort ranks via counting ----
        for (int i = tid; i < D; i += NTHREADS) {
            const float gi = g_s[i];
            int cnt = 0;
            for (int j = 0; j < D; ++j) {
                const float gj = g_s[j];
                cnt += (gj > gi) || (gj == gi && j < i);
            }
            r_s[i] = cnt;
        }
        if (tid == 0) flag_s = 0xFF;
        __syncthreads();

        // ---- Phase C: duplicate-vertex check against active rows ----
        {
            int m = 0xFF;
            #pragma unroll
            for (int k = 0; k < K; ++k) {
                if (!act_s[k]) continue;
                bool ok = true;
                for (int i = tid; i < D; i += NTHREADS)
                    ok = ok && (Rg[k * D + i] == r_s[i]);
                if (!ok) m &= ~(1 << k);
            }
            if (m != 0xFF) atomicAnd(&flag_s, m);
        }
        __syncthreads();
        if (tid == 0) {
            int dup = 0;
            #pragma unroll
            for (int k = 0; k < K; ++k)
                if (act_s[k] && ((flag_s >> k) & 1)) dup = 1;
            add_s = !dup;
            if (!dup) act_s[t] = 1;
            #pragma unroll
            for (int k = 0; k < K; ++k) a_s[k] = act_s[k] ? 1.f : 0.f;
        }
        __syncthreads();

        // ---- Phase D: install new vertex in slot t ----
        if (add_s) {
            for (int i = tid; i < D; i += NTHREADS) {
                const int r = r_s[i];
                Vg[t * D + i] = (float)(D - 1 - r);
                Rg[t * D + i] = r;
            }
        }
        for (int i = tid; i < 256; i += NTHREADS) G_s[i] = 0.f;
        __threadfence();
        __syncthreads();

        // ---- Phase E: fused Gram+b via f32 WMMA (16x16, K-inner = D) ----
        // W rows: 0..7 = a_k * V_k, row 8 = th, rows 9..15 = 0.
        // G_s[i][j] = Va_i . Va_j (i,j<8);  G_s[i][8] = b_i = a_i*(V_i.th)
        {
            const int lane = tid & 31;
            const int wave = tid >> 5;
            const int row  = lane & 15;          // matrix row this lane serves
            const int hi   = lane >> 4;          // K-half select (per ISA layout)
            v8f acc = (v8f)0.f;
            for (int c = wave * 4; c < D; c += NWAVES * 4) {
                const int col = c + 2 * hi;
                float w0, w1;
                if (row < K)       { const float a = a_s[row];
                                     w0 = a * Vg[row * D + col];
                                     w1 = a * Vg[row * D + col + 1]; }
                else if (row == K) { w0 = th_s[col]; w1 = th_s[col + 1]; }
                else               { w0 = 0.f;       w1 = 0.f; }
                v2f ab; ab[0] = w0; ab[1] = w1;
                // symmetric Gram: A and B lane-register contents are identical
                acc = __builtin_amdgcn_wmma_f32_16x16x4_f32(
                          false, ab, false, ab, (short)0, acc, false, false);
            }
            // C/D layout: acc[r] @ lane l -> (M = r + 8*hi, N = l&15)
            #pragma unroll
            for (int r = 0; r < 8; ++r)
                atomicAdd(&G_s[(r + 8 * hi) * 16 + row], acc[r]);
        }
        __syncthreads();

        // ---- Phase F/G: 9x9 KKT solve + line search (thread 0) ----
        if (tid == 0) {
            float A[9][9], rhs[9], x[9];
            #pragma unroll
            for (int i = 0; i < 8; ++i) {
                #pragma unroll
                for (int j = 0; j < 8; ++j) A[i][j] = G_s[i * 16 + j];
                const float a = a_s[i];
                A[i][i] += (1.f - a) + 1e-6f * a;
                A[i][8] = a; A[8][i] = a;
                rhs[i] = G_s[i * 16 + 8];
            }
            A[8][8] = 0.f; rhs[8] = 1.f;
            // Gaussian elimination with partial pivoting (matches LAPACK LU path)
            for (int c = 0; c < 9; ++c) {
                int p = c; float pv = fabsf(A[c][c]);
                for (int r = c + 1; r < 9; ++r) {
                    const float v = fabsf(A[r][c]);
                    if (v > pv) { pv = v; p = r; }
                }
                if (p != c) {
                    for (int j = 0; j < 9; ++j) { float tmp = A[c][j]; A[c][j] = A[p][j]; A[p][j] = tmp; }
                    float tb = rhs[c]; rhs[c] = rhs[p]; rhs[p] = tb;
                }
                const float inv = 1.f / A[c][c];
                for (int r = c + 1; r < 9; ++r) {
                    const float f = A[r][c] * inv;
                    for (int j = c; j < 9; ++j) A[r][j] -= f * A[c][j];
                    rhs[r] -= f * rhs[c];
                }
            }
            for (int c = 8; c >= 0; --c) {
                float s = rhs[c];
                for (int j = c + 1; j < 9; ++j) s -= A[c][j] * x[j];
                x[c] = s / A[c][c];
            }
            float qs[8];
            #pragma unroll
            for (int k = 0; k < 8; ++k) qs[k] = x[k] * a_s[k];
            float gamma = 1.f;
            #pragma unroll
            for (int k = 0; k < 8; ++k)
                if (qs[k] < 0.f && act_s[k])
                    gamma = fminf(gamma, q_s[k] / (q_s[k] - qs[k] + 1e-30f));
            float qn[8], ssum = 0.f;
            #pragma unroll
            for (int k = 0; k < 8; ++k) {
                qn[k] = q_s[k] + gamma * (qs[k] - q_s[k]);
                const int na = act_s[k] && (qn[k] > 1e-9f);
                act_s[k] = na;
                qn[k] = na ? qn[k] : 0.f;
                ssum += qn[k];
            }
            const float denom = fmaxf(ssum, 1e-30f);
            #pragma unroll
            for (int k = 0; k < 8; ++k) q_s[k] = qn[k] / denom;
        }
        __syncthreads();
    }

    // ---- finalize: alphas out; zero R rows of inactive slots ----
    if (tid < K) out[tid] = q_s[tid];
    #pragma unroll
    for (int k = 0; k < K; ++k)
        if (!act_s[k])
            for (int i = tid; i < D; i += NTHREADS) Rg[k * D + i] = 0;
}

// ---------------------------------------------------------------------------
// Kernel 2: mask expansion — 134 MB of streaming float4 non-temporal stores.
//   masks[k,i,j] = (R[k,i] < R[k,j]) ? 1 : 0.  R is 64KB -> fully L2-resident.
// ---------------------------------------------------------------------------
__global__ __launch_bounds__(256)
void mask_writer(const int* __restrict__ Rg, float* __restrict__ masks)
{
    const long long idx = (long long)blockIdx.x * blockDim.x + threadIdx.x;
    const int k   = (int)(idx >> 20);             // / (2048*512)
    const int rem = (int)(idx & ((1 << 20) - 1));
    const int i   = rem >> 9;                     // / 512
    const int j4  = (rem & 511) << 2;

    const int Ri = Rg[k * D + i];
    const v4i Rj = *(const v4i*)(Rg + k * D + j4);
    v4f m;
    m[0] = (Ri < Rj[0]) ? 1.f : 0.f;
    m[1] = (Ri < Rj[1]) ? 1.f : 0.f;
    m[2] = (Ri < Rj[2]) ? 1.f : 0.f;
    m[3] = (Ri < Rj[3]) ? 1.f : 0.f;

    float* dst = masks + ((long long)k * D * D + (long long)i * D + j4);
    __builtin_nontemporal_store(m, (v4f*)dst);
}

extern "C" void kernel_launch(void* const* d_in, const int* in_sizes, int n_in,
                              void* d_out, int out_size, void* d_ws, size_t ws_size,
                              hipStream_t stream) {
    const float* theta = (const float*)d_in[0];
    float* out = (float*)d_out;                 // [0..7]=alphas, then masks
    float* Vg = (float*)d_ws;                   // 8*2048 f32
    int*   Rg = (int*)((char*)d_ws + (size_t)K * D * sizeof(float));

    sparse_rank_solver<<<1, NTHREADS, 0, stream>>>(theta, out, Vg, Rg);

    const long long total_v4 = (long long)K * D * (D / 4);  // 8,388,608
    mask_writer<<<(int)(total_v4 / 256), 256, 0, stream>>>(Rg, out + K);
}